// WeightedCrossEntropy_42021960024747
// MI455X (gfx1250) — compile-verified
//
#include <hip/hip_runtime.h>
#include <hip/hip_bf16.h>
#include <stdint.h>
#include <math.h>

namespace {
constexpr int kN      = 4;
constexpr int kC      = 151;
constexpr int kHW     = 512 * 512;          // 2^18
constexpr int kPix    = kN * kHW;           // 1,048,576
constexpr int kBlock  = 256;
constexpr int kChunk  = 16;                 // channels staged per async tile
constexpr int kNChunks = 9;                 // 9*16 = 144 channels via async LDS
constexpr int kTailC  = kNChunks * kChunk;  // 144 .. 150 handled scalar
constexpr int kOldCl  = 100;
constexpr int kIgnore = 255;
constexpr int kBlocks = kPix / kBlock;      // 4096
}

__global__ __launch_bounds__(kBlock)
void wce_main_kernel(const float* __restrict__ x, const int* __restrict__ tgt,
                     const float* __restrict__ snp, float* __restrict__ partial)
{
    __shared__ float buf[2][kChunk * kBlock];   // 32 KB double buffer
    __shared__ float red[kBlock];

    const int t   = threadIdx.x;
    const int p0  = blockIdx.x * kBlock;
    const int n   = p0 >> 18;                   // / kHW
    const int hw0 = p0 & (kHW - 1);
    const int hw  = hw0 + t;

    const int rowSel = t >> 6;                  // 0..3
    const int kOff   = (t & 63) * 4;            // float index in 256-px row

    // per-pixel scalar inputs
    const int   tg    = tgt[(size_t)n * kHW + hw];
    const bool  valid = (tg != kIgnore);
    const int   mt    = valid ? tg : 0;
    const int   lab2  = (tg < kOldCl) ? 0 : tg;             // 255 stays 255
    const int   lab   = lab2 > (kC - 1) ? (kC - 1) : lab2;  // clip to [0,150]
    const float pa    = snp[((size_t)n * 2 + 0) * kHW + hw];
    const float pb    = snp[((size_t)n * 2 + 1) * kHW + hw];
    float msn = fmaxf(pa, pb);
    msn = (msn > 0.5f) ? 1.0f : msn;

    // streaming logsumexp state
    float m = -INFINITY, s = 0.0f, sOld = 0.0f, x0 = 0.0f, xt = 0.0f;

    auto issue = [&](int ci, int b) {
        const int c0 = ci * kChunk;
#pragma unroll
        for (int r = 0; r < 4; ++r) {
            const int j = r * 4 + rowSel;       // channel row within chunk
            const unsigned long long ga = (unsigned long long)(uintptr_t)
                (x + ((size_t)(n * kC + c0 + j) * kHW + (hw0 + kOff)));
            const unsigned lds = (unsigned)(uintptr_t)(&buf[b][j * kBlock + kOff]);
            asm volatile("global_load_async_to_lds_b128 %0, %1, off th:TH_LOAD_NT"
                         :: "v"(lds), "v"(ga) : "memory");
        }
    };

    issue(0, 0);
    for (int ci = 0; ci < kNChunks; ++ci) {
        asm volatile("s_wait_asynccnt 0x0" ::: "memory"); // my chunk-ci loads landed
        __syncthreads();                                   // everyone's landed; prev reads done
        if (ci + 1 < kNChunks) issue(ci + 1, (ci + 1) & 1); // overlap with compute below

        const int c0 = ci * kChunk;
        const float* bp = &buf[ci & 1][0];
        float v[kChunk];
#pragma unroll
        for (int j = 0; j < kChunk; ++j) v[j] = bp[j * kBlock + t];

        float cmax = v[0];
#pragma unroll
        for (int j = 1; j < kChunk; ++j) cmax = fmaxf(cmax, v[j]);
        const float nm = fmaxf(m, cmax);
        const float sc = __expf(m - nm);        // exp(-inf)=0 on first chunk
        s *= sc; sOld *= sc; m = nm;
#pragma unroll
        for (int j = 0; j < kChunk; ++j) {
            const float e = __expf(v[j] - m);
            s += e;
            sOld += ((c0 + j) < kOldCl) ? e : 0.0f;
            x0 = ((c0 + j) == 0)  ? v[j] : x0;
            xt = ((c0 + j) == lab) ? v[j] : xt;
        }
    }

    // tail channels 144..150, plain NT loads
#pragma unroll
    for (int c = kTailC; c < kC; ++c) {
        const float v = __builtin_nontemporal_load(
            x + ((size_t)(n * kC + c) * kHW + hw));
        const float nm = fmaxf(m, v);
        const float sc = __expf(m - nm);
        s = s * sc + __expf(v - nm);
        sOld *= sc;
        m = nm;
        xt = (c == lab) ? v : xt;
    }

    // per-pixel loss
    const float den   = m + __logf(s);
    const float outBg = x0 - den;
    const float outFg = m + __logf(s - __expf(x0 - m)) - den;   // LSE(c>=1) - den
    const float w     = (mt == 0) ? msn : 0.0f;
    const float focal = (1.0f - w) * (1.0f - w);
    const float picked = (mt == 0) ? outBg : outFg;
    const float l1 = valid ? (-picked) * focal : 0.0f;
    const float out0n = m + __logf(sOld) - den;                 // LSE(c<100) - den
    const float val = (lab == 0) ? out0n : (xt - den);
    const float l2 = valid ? -val : 0.0f;                       // valid2 == valid

    red[t] = l1 + l2;
    __syncthreads();
#pragma unroll
    for (int off = kBlock / 2; off > 0; off >>= 1) {
        if (t < off) red[t] += red[t + off];
        __syncthreads();
    }
    if (t == 0) partial[blockIdx.x] = red[0];
}

__global__ __launch_bounds__(kBlock)
void wce_reduce_kernel(const float* __restrict__ partial, float* __restrict__ out)
{
    __shared__ float red[kBlock];
    float a = 0.0f;
    for (int i = threadIdx.x; i < kBlocks; i += kBlock) a += partial[i];
    red[threadIdx.x] = a;
    __syncthreads();
#pragma unroll
    for (int off = kBlock / 2; off > 0; off >>= 1) {
        if (threadIdx.x < off) red[threadIdx.x] += red[threadIdx.x + off];
        __syncthreads();
    }
    if (threadIdx.x == 0) out[0] = red[0] * (1.0f / (float)kPix);
}

extern "C" void kernel_launch(void* const* d_in, const int* in_sizes, int n_in,
                              void* d_out, int out_size, void* d_ws, size_t ws_size,
                              hipStream_t stream)
{
    const float* x   = (const float*)d_in[0];   // [N,C,H,W] f32 logits
    const int*   tg  = (const int*)d_in[1];     // [N,H,W] i32 targets
    const float* snp = (const float*)d_in[2];   // [N,2,H,W] f32 probs
    (void)in_sizes; (void)n_in; (void)out_size; (void)ws_size;
    float* partial = (float*)d_ws;              // 4096 floats = 16 KB scratch

    wce_main_kernel<<<kBlocks, kBlock, 0, stream>>>(x, tg, snp, partial);
    wce_reduce_kernel<<<1, kBlock, 0, stream>>>(partial, (float*)d_out);
}